// PIFNode_25812753449785
// MI455X (gfx1250) — compile-verified
//
#include <hip/hip_runtime.h>

// PIF integrate-and-fire, T=8, threshold=1.0, v_init=0.5, fp32.
//
// Bit-exact closed form of the reference scan:
//   s = clamp(floor(x + 0.5f), 0.0f, 8.0f)
// Proof sketch: while v >= 1.0, fp32 (v - 1.0f) is exact (Sterbenz on [1,2),
// ulp argument for v >= 2), so v_t == v0 - t exactly for every step that
// spikes; after the first non-spike step all later comparisons are false.
//
// Memory-bound: 411 MB of HBM traffic -> ~17.6 us floor at 23.3 TB/s.
// Working set (2 x 205 MB) exceeds the 192 MB L2, so we stream with
// non-temporal b128 loads/stores (CDNA5 TH=NT cache hints).

typedef __attribute__((ext_vector_type(4))) float v4f;

namespace {

constexpr float kThresh      = 1.0f;   // V_THRESHOLD
constexpr float kVInit       = 0.5f;   // V_INIT
constexpr float kTSteps      = 8.0f;   // T
constexpr int   kBlock       = 256;    // 8 wave32 waves
constexpr int   kV4PerThread = 4;      // 64 B per thread
constexpr unsigned kPerBlock = kBlock * kV4PerThread;  // float4s per block

__device__ __forceinline__ float pif_count(float x) {
  float s = __builtin_floorf(x + kVInit);
  s = __builtin_fmaxf(s, 0.0f);
  s = __builtin_fminf(s, kTSteps);   // min/max clamp -> v_med3-style codegen
  return s;
}

__device__ __forceinline__ v4f pif_count4(v4f v) {
  v4f s;
  s.x = pif_count(v.x);
  s.y = pif_count(v.y);
  s.z = pif_count(v.z);
  s.w = pif_count(v.w);
  return s;
}

__global__ __launch_bounds__(kBlock) void pif_v4_kernel(
    const v4f* __restrict__ x, v4f* __restrict__ out, unsigned n4) {
  const unsigned tileBase = blockIdx.x * kPerBlock;
  const unsigned base     = tileBase + threadIdx.x;

  if (tileBase + kPerBlock <= n4) {
    // Fast path: whole tile in range. 4 outstanding NT b128 loads, then
    // VALU, then 4 NT b128 stores.
    v4f v[kV4PerThread];
#pragma unroll
    for (int i = 0; i < kV4PerThread; ++i)
      v[i] = __builtin_nontemporal_load(&x[base + i * kBlock]);
#pragma unroll
    for (int i = 0; i < kV4PerThread; ++i)
      __builtin_nontemporal_store(pif_count4(v[i]), &out[base + i * kBlock]);
  } else {
    // Guarded tail tile (not hit for the reference shape, kept for safety).
#pragma unroll
    for (int i = 0; i < kV4PerThread; ++i) {
      const unsigned idx = base + i * kBlock;
      if (idx < n4) {
        v4f v = __builtin_nontemporal_load(&x[idx]);
        __builtin_nontemporal_store(pif_count4(v), &out[idx]);
      }
    }
  }
}

// Handles n % 4 != 0 leftovers (never triggered for the reference shape).
__global__ void pif_tail_kernel(const float* __restrict__ x,
                                float* __restrict__ out,
                                unsigned start, unsigned n) {
  const unsigned i = start + blockIdx.x * blockDim.x + threadIdx.x;
  if (i < n) out[i] = pif_count(x[i]);
}

}  // namespace

extern "C" void kernel_launch(void* const* d_in, const int* in_sizes, int n_in,
                              void* d_out, int out_size, void* d_ws, size_t ws_size,
                              hipStream_t stream) {
  (void)n_in; (void)out_size; (void)d_ws; (void)ws_size;

  const float* x  = (const float*)d_in[0];
  float* out      = (float*)d_out;
  const unsigned n    = (unsigned)in_sizes[0];
  const unsigned n4   = n / 4u;
  const unsigned tail = n - n4 * 4u;

  if (n4) {
    const unsigned blocks = (n4 + kPerBlock - 1u) / kPerBlock;  // 12,544 here
    pif_v4_kernel<<<dim3(blocks), dim3(kBlock), 0, stream>>>(
        (const v4f*)x, (v4f*)out, n4);
  }
  if (tail) {
    pif_tail_kernel<<<dim3(1), dim3(64), 0, stream>>>(x, out, n4 * 4u, n);
  }
}